// FusionNetTVRN_51513837748720
// MI455X (gfx1250) — compile-verified
//
#include <hip/hip_runtime.h>

// Softmax-splatting ("average") forward warp, LDS-privatized scatter.
// - bilinear splats accumulate in a padded per-tile LDS buffer (ds_add_f32)
// - nonzero cells flushed with non-returning global_atomic_add_f32
// - rare far splats (|disp| > R) fall back to direct global atomics
// - streaming passes vectorized float4 with NT cache hints
// All element offsets < 2^31 -> pure 32-bit index math.

namespace {

typedef float v4f __attribute__((ext_vector_type(4)));

constexpr int N   = 8;
constexpr int H   = 720;
constexpr int W   = 1280;
constexpr int HW  = H * W;              // 921600
constexpr int NHW = N * HW;             // 7372800
constexpr int BLK = 256;                // 8 wave32 waves per block

// Tiling for the splat kernel
constexpr int TW = 64, TH = 16, R = 4;  // tile + apron radius
constexpr int PW = TW + 2 * R;          // 72
constexpr int PH = TH + 2 * R;          // 24
constexpr int CELLS = PW * PH;          // 1728 cells -> 27648 B LDS
constexpr int PXPT  = (TW * TH) / BLK;  // 4 pixels per thread

// d_out layout (floats):
//   out0 (concat, 6 ch/batch): plane (n*6 + c)
//   out1 (warped_img1)       : plane (6N + n*3 + c)
//   out2 (warped_img0)       : plane (9N + n*3 + c)
// During accumulation: den1[n] = out0 plane (6n+0), den0[n] = out0 plane (6n+1)

__device__ __forceinline__ void atom_add_f32(float* p, float v) {
  // relaxed, device scope, result unused -> non-returning global_atomic_add_f32
  (void)__hip_atomic_fetch_add(p, v, __ATOMIC_RELAXED, __HIP_MEMORY_SCOPE_AGENT);
}

__device__ __forceinline__ void lds_add(float* p, float v) {
  // workgroup scope, result unused -> non-returning ds_add_f32
  (void)__hip_atomic_fetch_add(p, v, __ATOMIC_RELAXED, __HIP_MEMORY_SCOPE_WORKGROUP);
}

// ---- Stage 0a: zero the contiguous numerator region [6N*HW, 12N*HW) --------
__global__ __launch_bounds__(BLK) void zero_contig_kernel(v4f* __restrict__ p) {
  int i = blockIdx.x * BLK + threadIdx.x;
  const int n4 = (6 * N * HW) / 4;
  if (i < n4) p[i] = (v4f)0.0f;
}

// ---- Stage 0b: zero denominator planes (out0 planes 6n+0, 6n+1) ------------
__global__ __launch_bounds__(BLK) void zero_dens_kernel(float* __restrict__ out) {
  int i = blockIdx.x * BLK + threadIdx.x;
  const int n4 = (2 * N * HW) / 4;
  if (i >= n4) return;
  int e = i * 4;                    // element in [0, 2N*HW)
  int n = e / (2 * HW);
  int r = e - n * (2 * HW);         // [0, 2*HW): spans planes 6n+0, 6n+1
  *(v4f*)(out + n * 6 * HW + r) = (v4f)0.0f;
}

// ---- Stage 1: tiled bilinear forward splat (both images) -------------------
__global__ __launch_bounds__(BLK) void splat_tiled_kernel(
    const float* __restrict__ img0, const float* __restrict__ img1,
    const float* __restrict__ flow, float* __restrict__ out) {
  __shared__ float acc[4 * CELLS];

  const int n   = blockIdx.z;
  const int tx0 = blockIdx.x * TW;
  const int ty0 = blockIdx.y * TH;
  const int tid = threadIdx.x;
  const int lx  = tid & (TW - 1);   // 0..63 (lane-consecutive x: LDS stride-1)
  const int ly0 = tid >> 6;         // 0..3  (rows ly0, ly0+4, ly0+8, ly0+12)
  const int x   = tx0 + lx;

  // preload flow for this thread's 4 pixels (flow is consumed once -> NT)
  float fxv[2][PXPT], fyv[2][PXPT];
  const float* bf = flow + n * 4 * HW;
#pragma unroll
  for (int k = 0; k < PXPT; ++k) {
    int y = ty0 + ly0 + k * 4;
    int p = y * W + x;
    fxv[1][k] = __builtin_nontemporal_load(bf + p)          * 0.5f;  // img0 flow
    fyv[1][k] = __builtin_nontemporal_load(bf + HW + p)     * 0.5f;
    fxv[0][k] = __builtin_nontemporal_load(bf + 2 * HW + p) * 0.5f;  // img1 flow
    fyv[0][k] = __builtin_nontemporal_load(bf + 3 * HW + p) * 0.5f;
  }

  float* num1 = out + (6 * N + n * 3) * HW;
  float* num0 = out + (9 * N + n * 3) * HW;
  float* den1 = out + (n * 6 + 0) * HW;
  float* den0 = out + (n * 6 + 1) * HW;

  for (int s = 0; s < 2; ++s) {
    const float* img  = (s == 0) ? img1 : img0;
    float*       numg = (s == 0) ? num1 : num0;
    float*       deng = (s == 0) ? den1 : den0;

    // zero the LDS tile
    for (int i = tid; i < 4 * CELLS; i += BLK) acc[i] = 0.0f;
    __syncthreads();

    // splat this thread's pixels into LDS (fallback: direct global atomics)
#pragma unroll
    for (int k = 0; k < PXPT; ++k) {
      int y = ty0 + ly0 + k * 4;
      int p = y * W + x;
      const float* ip = img + n * 3 * HW + p;
      float v0 = ip[0], v1 = ip[HW], v2 = ip[2 * HW];

      float gx  = (float)x + fxv[s][k];
      float gy  = (float)y + fyv[s][k];
      float x0f = floorf(gx), y0f = floorf(gy);
      float ftx = gx - x0f,   fty = gy - y0f;
      int   x0  = (int)x0f,   y0  = (int)y0f;
      float wx[2] = {1.0f - ftx, ftx};
      float wy[2] = {1.0f - fty, fty};
#pragma unroll
      for (int dy = 0; dy < 2; ++dy) {
#pragma unroll
        for (int dx = 0; dx < 2; ++dx) {
          int xi = x0 + dx, yi = y0 + dy;
          if ((unsigned)xi >= (unsigned)W || (unsigned)yi >= (unsigned)H) continue;
          float w = wx[dx] * wy[dy];
          int cx = xi - tx0 + R, cy = yi - ty0 + R;
          if ((unsigned)cx < (unsigned)PW && (unsigned)cy < (unsigned)PH) {
            int ci = cy * PW + cx;
            lds_add(&acc[ci],             v0 * w);
            lds_add(&acc[CELLS + ci],     v1 * w);
            lds_add(&acc[2 * CELLS + ci], v2 * w);
            lds_add(&acc[3 * CELLS + ci], w);
          } else {  // rare: |displacement| > R
            int gi = yi * W + xi;
            atom_add_f32(numg + gi,          v0 * w);
            atom_add_f32(numg + gi + HW,     v1 * w);
            atom_add_f32(numg + gi + 2 * HW, v2 * w);
            atom_add_f32(deng + gi,          w);
          }
        }
      }
    }
    __syncthreads();

    // flush nonzero LDS cells (plane-major: plane index is loop-constant)
#pragma unroll
    for (int q = 0; q < 4; ++q) {
      float* base = (q < 3) ? (numg + q * HW) : deng;
      const float* lp = acc + q * CELLS;
      for (int i = tid; i < CELLS; i += BLK) {
        float v = lp[i];
        if (v == 0.0f) continue;
        int cy  = i / PW;                 // const-div -> mul/shift
        int cx  = i - cy * PW;
        int gxp = tx0 - R + cx;
        int gyp = ty0 - R + cy;
        if ((unsigned)gxp < (unsigned)W && (unsigned)gyp < (unsigned)H)
          atom_add_f32(base + gyp * W + gxp, v);
      }
    }
    __syncthreads();  // protect LDS before next phase's zeroing
  }
}

// ---- Stage 2: normalize warped images in place; write out0 ch3..5 ----------
__global__ __launch_bounds__(BLK) void norm_kernel(const float* __restrict__ img0,
                                                   const float* __restrict__ img1,
                                                   float* __restrict__ out) {
  int g = blockIdx.x * BLK + threadIdx.x;   // over NHW/4
  if (g >= NHW / 4) return;
  int n = g / (HW / 4);
  int p = (g - n * (HW / 4)) * 4;

  v4f d1 = *(const v4f*)(out + (n * 6 + 0) * HW + p);
  v4f d0 = *(const v4f*)(out + (n * 6 + 1) * HW + p);
  v4f r1, r0;
#pragma unroll
  for (int j = 0; j < 4; ++j) {
    r1[j] = 1.0f / ((d1[j] == 0.0f) ? 1.0f : d1[j]);
    r0[j] = 1.0f / ((d0[j] == 0.0f) ? 1.0f : d0[j]);
  }

  float* o1 = out + (6 * N + n * 3) * HW + p;
  float* o0 = out + (9 * N + n * 3) * HW + p;
#pragma unroll
  for (int c = 0; c < 3; ++c) {
    *(v4f*)(o1 + c * HW) = *(const v4f*)(o1 + c * HW) * r1;
    *(v4f*)(o0 + c * HW) = *(const v4f*)(o0 + c * HW) * r0;
  }

  // out0 ch3..5 = img1 - img0 (write-once, never read back -> NT store)
  const float* i0 = img0 + n * 3 * HW + p;
  const float* i1 = img1 + n * 3 * HW + p;
  float* oc = out + (n * 6 + 3) * HW + p;
#pragma unroll
  for (int c = 0; c < 3; ++c) {
    v4f a = *(const v4f*)(i1 + c * HW);
    v4f b = *(const v4f*)(i0 + c * HW);
    __builtin_nontemporal_store(a - b, (v4f*)(oc + c * HW));
  }
}

// ---- Stage 3: out0 ch0..2 = warped1 - warped0 (overwrites dead dens) -------
__global__ __launch_bounds__(BLK) void diff_kernel(float* __restrict__ out) {
  int g = blockIdx.x * BLK + threadIdx.x;   // over NHW/4
  if (g >= NHW / 4) return;
  int n = g / (HW / 4);
  int p = (g - n * (HW / 4)) * 4;

  const float* o1 = out + (6 * N + n * 3) * HW + p;
  const float* o0 = out + (9 * N + n * 3) * HW + p;
  float* d = out + (n * 6) * HW + p;
#pragma unroll
  for (int c = 0; c < 3; ++c) {
    v4f a = *(const v4f*)(o1 + c * HW);
    v4f b = *(const v4f*)(o0 + c * HW);
    __builtin_nontemporal_store(a - b, (v4f*)(d + c * HW));
  }
}

}  // namespace

extern "C" void kernel_launch(void* const* d_in, const int* in_sizes, int n_in,
                              void* d_out, int out_size, void* d_ws, size_t ws_size,
                              hipStream_t stream) {
  (void)in_sizes; (void)n_in; (void)out_size; (void)d_ws; (void)ws_size;
  const float* img0 = (const float*)d_in[0];
  const float* img1 = (const float*)d_in[1];
  const float* flow = (const float*)d_in[2];
  float* out = (float*)d_out;

  // Stage 0: zero accumulators (numerators out1/out2 + dens in out0 ch0/1)
  {
    int n4 = (6 * N * HW) / 4;
    zero_contig_kernel<<<(n4 + BLK - 1) / BLK, BLK, 0, stream>>>(
        (v4f*)(out + 6 * N * HW));
  }
  {
    int n4 = (2 * N * HW) / 4;
    zero_dens_kernel<<<(n4 + BLK - 1) / BLK, BLK, 0, stream>>>(out);
  }

  // Stage 1: LDS-privatized scatter-add splat
  {
    dim3 grid(W / TW, H / TH, N);  // 20 x 45 x 8 = 7200 blocks
    splat_tiled_kernel<<<grid, BLK, 0, stream>>>(img0, img1, flow, out);
  }

  int blocks4 = (NHW / 4 + BLK - 1) / BLK;

  // Stage 2: normalize + img diff channels (float4)
  norm_kernel<<<blocks4, BLK, 0, stream>>>(img0, img1, out);

  // Stage 3: warped diff channels (float4)
  diff_kernel<<<blocks4, BLK, 0, stream>>>(out);
}